// ModelNew_23656679867212
// MI455X (gfx1250) — compile-verified
//
#include <hip/hip_runtime.h>

// MI455X (gfx1250, wave32) fused kernel for:
//   y = x @ W^T + b ; pooled = maxpool1d(y, k=4, s=4) ; out = 0.5 * pooled.sum(-1)
//
// Roofline: 2*4096^3 = 137 GFLOP vs 128 MB mandatory traffic (~5.5us @ 23.3TB/s)
// -> compute bound on the f32 matrix path. Inputs/accum are fp32, so we use the
// exact-precision V_WMMA_F32_16X16X4_F32. Epilogue (bias, maxpool4, sum, *0.5)
// is fused in-register so the 64MB intermediate never exists.
//
//  - 256x128 workgroup tile, 8 waves as 4Mx2N, each wave owns 64x64 = 4x4
//    WMMA tiles -> 16 WMMAs per 8 fragment b64 loads.
//  - LDS stride pad +4 (36 dwords): conflict-free b64 frag reads AND 16B-aligned
//    rows for b128 LDS writes.
//  - GLOBAL_LOAD_ASYNC_TO_LDS_B128 double-buffered pipeline (ASYNCcnt), one
//    barrier per K-slab, no VGPR staging. The builtin takes
//    (addrspace(1) int4*, addrspace(3) int4*, imm offset, imm cpol).

typedef __attribute__((ext_vector_type(2))) float v2f;
typedef __attribute__((ext_vector_type(4))) float v4f;
typedef __attribute__((ext_vector_type(8))) float v8f;
typedef int v4i_gcc __attribute__((vector_size(16)));

#define N_FEAT 4096
#define TILE_M 256
#define TILE_N 128
#define TILE_K 32
#define LDS_STRIDE (TILE_K + 4) // 36 dwords: conflict-free + 16B-aligned rows

#if defined(__has_builtin)
#if __has_builtin(__builtin_amdgcn_global_load_async_to_lds_b128) && \
    __has_builtin(__builtin_amdgcn_s_wait_asynccnt)
#define USE_ASYNC_LDS 1
#endif
#endif
#ifndef USE_ASYNC_LDS
#define USE_ASYNC_LDS 0
#endif

#if USE_ASYNC_LDS
#define NBUF 2
#define GLOBAL_AS(p) ((__attribute__((address_space(1))) v4i_gcc*)(p))
#define LDS_AS(p)    ((__attribute__((address_space(3))) v4i_gcc*)(p))
#else
#define NBUF 1
#endif

__global__ void zero_out_kernel(float* __restrict__ out, int n) {
    int i = blockIdx.x * blockDim.x + threadIdx.x;
    if (i < n) out[i] = 0.0f;
}

__global__ __launch_bounds__(256) void fused_gemm_pool_kernel(
    const float* __restrict__ x,   // [4096, 4096] row-major
    const float* __restrict__ W,   // [4096, 4096] row-major ([o][i])
    const float* __restrict__ bias,// [4096]
    float* __restrict__ out)       // [4096], pre-zeroed
{
    __shared__ __align__(16) float As[NBUF][TILE_M * LDS_STRIDE]; // x tile [m][k]
    __shared__ __align__(16) float Ws[NBUF][TILE_N * LDS_STRIDE]; // W tile [o][k]

    const int tid   = threadIdx.x;
    const int lane  = tid & 31;
    const int wave  = tid >> 5;     // 0..7
    const int waveM = wave >> 1;    // 0..3 -> 64-row slab
    const int waveN = wave & 1;     // 0..1 -> 64-col slab
    const int half  = lane >> 4;    // 0/1: K-pair select in A/B frags, M-half in C/D
    const int l16   = lane & 15;

    const int m0 = blockIdx.y * TILE_M;
    const int n0 = blockIdx.x * TILE_N;

    // Staging map: thread -> (row within 32-row pass, 4-dword column chunk)
    const int lr = tid >> 3;        // 0..31
    const int lc = (tid & 7) * 4;   // 0..28 (8 threads x 16B = 128B/row segment)

    v8f acc[4][4];
    {
        v8f z = {};
        #pragma unroll
        for (int i = 0; i < 4; ++i)
            #pragma unroll
            for (int j = 0; j < 4; ++j) acc[i][j] = z;
    }

    // Fragment base offsets (lane-dependent, buffer-independent).
    // A frag (16x4 f32): lane l holds A[M=l%16][K = 2*half + {0,1}] -> one b64.
    // B frag (4x16 f32): lane l holds B[K = 2*half + {0,1}][N=l%16]; Ws stored
    // [o][k] makes this two consecutive dwords -> one b64.
    const int aOff = (waveM * 64 + l16) * LDS_STRIDE + 2 * half;
    const int bOff = (waveN * 64 + l16) * LDS_STRIDE + 2 * half;

    int buf = 0;

#if USE_ASYNC_LDS
    // ---- prologue: async-stage slab 0 into buffer 0
    {
        #pragma unroll
        for (int p = 0; p < TILE_M / 32; ++p) {
            int r = p * 32 + lr;
            __builtin_amdgcn_global_load_async_to_lds_b128(
                GLOBAL_AS(x + (size_t)(m0 + r) * N_FEAT + lc),
                LDS_AS(&As[0][r * LDS_STRIDE + lc]), 0, 0);
        }
        #pragma unroll
        for (int p = 0; p < TILE_N / 32; ++p) {
            int r = p * 32 + lr;
            __builtin_amdgcn_global_load_async_to_lds_b128(
                GLOBAL_AS(W + (size_t)(n0 + r) * N_FEAT + lc),
                LDS_AS(&Ws[0][r * LDS_STRIDE + lc]), 0, 0);
        }
        __builtin_amdgcn_s_wait_asynccnt(0);
        __syncthreads();
    }
#endif

    for (int kb = 0; kb < N_FEAT; kb += TILE_K) {
        const bool has_next = (kb + TILE_K) < N_FEAT;

#if USE_ASYNC_LDS
        // ---- async-stage next slab into the other buffer (no VGPR staging)
        if (has_next) {
            const int nb = buf ^ 1;
            const int kn = kb + TILE_K;
            #pragma unroll
            for (int p = 0; p < TILE_M / 32; ++p) {
                int r = p * 32 + lr;
                __builtin_amdgcn_global_load_async_to_lds_b128(
                    GLOBAL_AS(x + (size_t)(m0 + r) * N_FEAT + kn + lc),
                    LDS_AS(&As[nb][r * LDS_STRIDE + lc]), 0, 0);
            }
            #pragma unroll
            for (int p = 0; p < TILE_N / 32; ++p) {
                int r = p * 32 + lr;
                __builtin_amdgcn_global_load_async_to_lds_b128(
                    GLOBAL_AS(W + (size_t)(n0 + r) * N_FEAT + kn + lc),
                    LDS_AS(&Ws[nb][r * LDS_STRIDE + lc]), 0, 0);
            }
        }
#else
        // ---- sync staging into buffer 0
        #pragma unroll
        for (int p = 0; p < TILE_M / 32; ++p) {
            int r = p * 32 + lr;
            *(v4f*)&As[0][r * LDS_STRIDE + lc] =
                *(const v4f*)(x + (size_t)(m0 + r) * N_FEAT + kb + lc);
        }
        #pragma unroll
        for (int p = 0; p < TILE_N / 32; ++p) {
            int r = p * 32 + lr;
            *(v4f*)&Ws[0][r * LDS_STRIDE + lc] =
                *(const v4f*)(W + (size_t)(n0 + r) * N_FEAT + kb + lc);
        }
        if (has_next) { // gfx1250 cache prefetch of next K-slab
            __builtin_prefetch(x + (size_t)(m0 + lr) * N_FEAT + kb + TILE_K + lc, 0, 1);
            __builtin_prefetch(W + (size_t)(n0 + (lr & (TILE_N - 1))) * N_FEAT + kb + TILE_K + lc, 0, 1);
        }
        __syncthreads();
#endif

        // ---- compute: 8 K-steps of 4; 4x4 tiles -> 16 WMMAs per 8 frag loads
        {
            const float* pa = &As[buf][aOff];
            const float* pb = &Ws[buf][bOff];
            #pragma unroll
            for (int kk = 0; kk < TILE_K; kk += 4) {
                v2f a[4], b[4];
                #pragma unroll
                for (int t = 0; t < 4; ++t) {
                    a[t] = *(const v2f*)(pa + t * 16 * LDS_STRIDE + kk);
                    b[t] = *(const v2f*)(pb + t * 16 * LDS_STRIDE + kk);
                }
                #pragma unroll
                for (int tm = 0; tm < 4; ++tm)
                    #pragma unroll
                    for (int tn = 0; tn < 4; ++tn)
                        acc[tm][tn] = __builtin_amdgcn_wmma_f32_16x16x4_f32(
                            false, a[tm], false, b[tn], (short)0,
                            acc[tm][tn], false, false);
            }
        }

#if USE_ASYNC_LDS
        if (has_next) __builtin_amdgcn_s_wait_asynccnt(0);
        if (has_next) __syncthreads(); // next slab fully resident before flip
        buf ^= 1;
#else
        __syncthreads();
#endif
    }

    // ---- fused epilogue: +bias, maxpool(4) along n, sum windows, atomic row add
    // C/D layout: lane l, component v holds y[m = base + v + 8*half][n = nb + l16].
    // Pool windows of 4 == lane groups {4j..4j+3} within each 16-lane half.
    float bv[4];
    #pragma unroll
    for (int tn = 0; tn < 4; ++tn)
        bv[tn] = bias[n0 + waveN * 64 + tn * 16 + l16];

    #pragma unroll
    for (int tm = 0; tm < 4; ++tm) {
        #pragma unroll
        for (int v = 0; v < 8; ++v) {
            float rowsum = 0.0f;
            #pragma unroll
            for (int tn = 0; tn < 4; ++tn) {
                float val = acc[tm][tn][v] + bv[tn];
                // max over the 4-lane pooling window
                val = fmaxf(val, __shfl_xor(val, 1, 32));
                val = fmaxf(val, __shfl_xor(val, 2, 32));
                // one representative per window, then sum 4 windows per half
                val = (lane & 3) ? 0.0f : val;
                val += __shfl_xor(val, 4, 32);
                val += __shfl_xor(val, 8, 32);
                rowsum += val;
            }
            if (l16 == 0) { // lanes 0 and 16 -> rows v and v+8
                int m = m0 + waveM * 64 + tm * 16 + v + 8 * half;
                atomicAdd(out + m, 0.5f * rowsum);
            }
        }
    }
}

extern "C" void kernel_launch(void* const* d_in, const int* in_sizes, int n_in,
                              void* d_out, int out_size, void* d_ws, size_t ws_size,
                              hipStream_t stream) {
    (void)in_sizes; (void)n_in; (void)d_ws; (void)ws_size;
    const float* x = (const float*)d_in[0];
    const float* W = (const float*)d_in[1];
    const float* b = (const float*)d_in[2];
    float* out = (float*)d_out;

    zero_out_kernel<<<(out_size + 255) / 256, 256, 0, stream>>>(out, out_size);

    dim3 grid(N_FEAT / TILE_N, N_FEAT / TILE_M); // 32 x 16 = 512 workgroups
    fused_gemm_pool_kernel<<<grid, 256, 0, stream>>>(x, W, b, out);
}